// CliffordKANLayer_71511205479239
// MI455X (gfx1250) — compile-verified
//
#include <hip/hip_runtime.h>
#include <hip/hip_bf16.h>
#include <stdint.h>

// Problem constants (from reference)
#define BN 2048     // batch
#define IN 128      // input dim
#define ON 128      // output dim
#define NN 256      // O*D columns
#define K_RBF 8192  // I*G*G
#define K_TOT 8448  // + I*D silu tail
#define KT_TOT (K_TOT/4)            // 2112 k-steps of 4
#define BF_FLOATS (KT_TOT*16*32*2)  // 2,162,688 floats in fragment layout

typedef __attribute__((ext_vector_type(2))) float v2f;
typedef __attribute__((ext_vector_type(8))) float v8f;

// ---------------------------------------------------------------------------
// Pack B matrix (8448 x 256) into WMMA-fragment-native layout:
//   bf[kt][nt][lane][c]  (c in {0,1})
//   lane<16 : B[4*kt +   c][16*nt + lane]
//   lane>=16: B[4*kt + 2+c][16*nt + lane-16]
// Rows k<8192: weights[i,o,u,v,z] with k=(i*64+uv), n=(o*2+z).
// Rows k>=8192: Cayley-folded silu_weight: sum_x sw[i,o,x]*C[x,y,z].
// ---------------------------------------------------------------------------
__global__ __launch_bounds__(256) void kan_pack(const float* __restrict__ w,
                                                const float* __restrict__ sw,
                                                const float* __restrict__ cay,
                                                float* __restrict__ bf) {
  unsigned tid = blockIdx.x * 256u + threadIdx.x;    // < BF_FLOATS exactly
  unsigned c    = tid & 1u;
  unsigned lane = (tid >> 1) & 31u;
  unsigned nt   = (tid >> 6) & 15u;
  unsigned kt   = tid >> 10;
  unsigned k = kt * 4u + ((lane >> 4) << 1) + c;
  unsigned n = nt * 16u + (lane & 15u);
  unsigned o = n >> 1, z = n & 1u;
  float val;
  if (k < K_RBF) {
    unsigned i = k >> 6, uv = k & 63u;
    // weights flat (I,O,G,G,D): (i*128+o)*128 + uv*2 + z
    val = w[((i << 7) + o) * 128u + (uv << 1) + z];
  } else {
    unsigned km = k - K_RBF;
    unsigned i = km >> 1, y = km & 1u;
    const float* swp = sw + (size_t)((i << 7) + o) * 2u;
    val = swp[0] * cay[(y << 1) + z] + swp[1] * cay[4u + (y << 1) + z];
  }
  bf[tid] = val;
}

// biassum[n] = sum_i silu_bias[i, o, z]
__global__ __launch_bounds__(256) void kan_bias(const float* __restrict__ sb,
                                                float* __restrict__ biassum) {
  int n = threadIdx.x;          // 256 threads
  int o = n >> 1, z = n & 1;
  float s = 0.f;
  for (int i = 0; i < IN; ++i) s += sb[(i * ON + o) * 2 + z];
  biassum[n] = s;
}

// ---------------------------------------------------------------------------
// Fused GEMM: out[b, oz] = sum_k A[b,k] * B[k, oz] + biassum[oz]
// A is computed on the fly in LDS (RBF blocks for k<8192, SiLU tail after),
// B blocks are staged in LDS via CDNA5 async-to-LDS loads.
// Grid: 32 blocks x 128 threads (4 waves); wave w owns rows [b0+16w, b0+16w+16).
// ---------------------------------------------------------------------------
__global__ __launch_bounds__(128) void kan_gemm(const float* __restrict__ x,
                                                const float* __restrict__ gridp,
                                                const float* __restrict__ bfrag,
                                                const float* __restrict__ biassum,
                                                float* __restrict__ out) {
  __shared__ __align__(16) float s_a[64][68];   // A tile, pad 4 -> conflict-free frag loads
  __shared__ __align__(16) float s_b[8192];     // 8 kt-steps of B fragments (32 KB)
  __shared__ __align__(16) float s_e[64][16];   // er[8] | ei[8] per row

  const int tid  = threadIdx.x;
  const int wave = tid >> 5;
  const int lane = tid & 31;
  const int b0   = blockIdx.x * 64;

  v8f acc[16];
  const v8f vzero = {0.f,0.f,0.f,0.f,0.f,0.f,0.f,0.f};
#pragma unroll
  for (int nt = 0; nt < 16; ++nt) acc[nt] = vzero;

  const unsigned sb_base = (unsigned)(uintptr_t)&s_b[0];   // LDS byte address (low 32 bits)
  const int arow = wave * 16 + (lane & 15);
  const int ksel = (lane >> 4) << 1;

  for (int blk = 0; blk < 132; ++blk) {
    // ---- build A tile (64 rows x 64 k-cols) in LDS ----
    if (blk < 128) {
      // RBF block for input channel i = blk (separable in u,v)
      const int r = tid & 63, half = tid >> 6;
      const float xv = x[(size_t)((b0 + r) * IN + blk) * 2 + half];
#pragma unroll
      for (int u = 0; u < 8; ++u) {
        const float g = half ? gridp[u * 2 + 1] : gridp[u * 16];
        const float d = xv - g;
        s_e[r][half * 8 + u] = __expf(-d * d);
      }
      __syncthreads();
      const int r2 = tid >> 1;
      const int c0 = (tid & 1) << 5;
#pragma unroll
      for (int c = 0; c < 32; ++c) {
        const int cc = c0 + c;
        s_a[r2][cc] = s_e[r2][cc >> 3] * s_e[r2][8 + (cc & 7)];
      }
    } else {
      // SiLU tail block: k = 8192 + (blk-128)*64 + cc  ->  (i = km/2, y = km&1)
      const int kb = (blk - 128) * 64;
      const int r2 = tid >> 1;
      const int c0 = (tid & 1) << 5;
#pragma unroll
      for (int c = 0; c < 32; ++c) {
        const int cc = c0 + c;
        const int km = kb + cc;
        const float xv = x[(size_t)((b0 + r2) * IN + (km >> 1)) * 2 + (km & 1)];
        s_a[r2][cc] = xv / (1.0f + __expf(-xv));
      }
    }
    __syncthreads();

    // ---- two half-blocks of 8 kt-steps each ----
    for (int h = 0; h < 2; ++h) {
      // async-stage 32 KB of B fragments into LDS
      const unsigned gbase = (unsigned)(blk * 16 + h * 8) * 4096u;  // bytes into bfrag
#pragma unroll
      for (int j = 0; j < 16; ++j) {
        const unsigned off  = (unsigned)(j * 2048 + tid * 16);
        const unsigned ldsa = sb_base + off;
        const unsigned goff = gbase + off;
        asm volatile("global_load_async_to_lds_b128 %0, %1, %2"
                     :: "v"(ldsa), "v"(goff), "s"(bfrag) : "memory");
      }
      asm volatile("s_wait_asynccnt 0" ::: "memory");
      __syncthreads();

      const float* sbp = &s_b[0];
      // Register-level software pipeline over kt: load kt+1's fragments while
      // kt's 16 WMMAs execute, so s_wait_dscnt lands between bursts.
      v2f a_cur;
      v2f b_cur[16];
      {
        const int cbase0 = ((h * 8) << 2) + ksel;
        a_cur = *(const v2f*)&s_a[arow][cbase0];
#pragma unroll
        for (int nt = 0; nt < 16; ++nt)
          b_cur[nt] = *(const v2f*)(sbp + (nt << 6) + (lane << 1));
      }
#pragma unroll
      for (int kt = 0; kt < 8; ++kt) {
        v2f a_nxt = {};
        v2f b_nxt[16];
        if (kt < 7) {
          const int cbase = ((h * 8 + kt + 1) << 2) + ksel;
          a_nxt = *(const v2f*)&s_a[arow][cbase];
          const float* bkt = sbp + ((kt + 1) << 10);
#pragma unroll
          for (int nt = 0; nt < 16; ++nt)
            b_nxt[nt] = *(const v2f*)(bkt + (nt << 6) + (lane << 1));
        }
#pragma unroll
        for (int nt = 0; nt < 16; ++nt)
          acc[nt] = __builtin_amdgcn_wmma_f32_16x16x4_f32(
              false, a_cur, false, b_cur[nt], (short)0, acc[nt], false, false);
        a_cur = a_nxt;
#pragma unroll
        for (int nt = 0; nt < 16; ++nt) b_cur[nt] = b_nxt[nt];
      }
      __syncthreads();   // protect s_b (and s_a/s_e on last h) before overwrite
    }
  }

  // ---- epilogue: C layout -> lanes 0-15: M=r, N=lane; lanes 16-31: M=8+r ----
  const int col = lane & 15;
  const int rowbase = b0 + wave * 16 + ((lane >> 4) << 3);
#pragma unroll
  for (int nt = 0; nt < 16; ++nt) {
    const int n = nt * 16 + col;
    const float bs = biassum[n];
#pragma unroll
    for (int r = 0; r < 8; ++r) {
      out[(size_t)(rowbase + r) * NN + n] = acc[nt][r] + bs;
    }
  }
}

// ---------------------------------------------------------------------------
// Deterministic BatchNorm: partial sums -> finalize -> apply (in place on out)
// ---------------------------------------------------------------------------
__global__ __launch_bounds__(256) void kan_red(const float* __restrict__ out,
                                               float* __restrict__ part) {
  __shared__ float ls[4][256];
  const int t = threadIdx.x;
  const size_t base = (size_t)blockIdx.x * 8192;
  float s0 = 0.f, s1 = 0.f, q0 = 0.f, q1 = 0.f;
#pragma unroll
  for (int j = 0; j < 32; ++j) {
    const float v = out[base + (size_t)j * 256 + t];
    if (t & 1) { s1 += v; q1 += v * v; } else { s0 += v; q0 += v * v; }
  }
  ls[0][t] = s0; ls[1][t] = s1; ls[2][t] = q0; ls[3][t] = q1;
  __syncthreads();
  for (int off = 128; off > 0; off >>= 1) {
    if (t < off) {
      ls[0][t] += ls[0][t + off];
      ls[1][t] += ls[1][t + off];
      ls[2][t] += ls[2][t + off];
      ls[3][t] += ls[3][t + off];
    }
    __syncthreads();
  }
  if (t == 0) {
    part[blockIdx.x * 4 + 0] = ls[0][0];
    part[blockIdx.x * 4 + 1] = ls[1][0];
    part[blockIdx.x * 4 + 2] = ls[2][0];
    part[blockIdx.x * 4 + 3] = ls[3][0];
  }
}

__global__ void kan_fin(const float* __restrict__ part,
                        const float* __restrict__ gamma,
                        const float* __restrict__ beta,
                        float* __restrict__ ab) {
  if (threadIdx.x == 0) {
    float s0 = 0.f, s1 = 0.f, q0 = 0.f, q1 = 0.f;
    for (int b = 0; b < 64; ++b) {
      s0 += part[b * 4 + 0]; s1 += part[b * 4 + 1];
      q0 += part[b * 4 + 2]; q1 += part[b * 4 + 3];
    }
    const float inv = 1.0f / 262144.0f;   // B*O per component
    const float m0 = s0 * inv, m1 = s1 * inv;
    const float v0 = q0 * inv - m0 * m0;
    const float v1 = q1 * inv - m1 * m1;
    const float a0 = gamma[0] * rsqrtf(v0 + 1e-5f);
    const float a1 = gamma[1] * rsqrtf(v1 + 1e-5f);
    ab[0] = a0; ab[1] = a1;
    ab[2] = beta[0] - m0 * a0;
    ab[3] = beta[1] - m1 * a1;
  }
}

__global__ __launch_bounds__(256) void kan_apply(float* __restrict__ out,
                                                 const float* __restrict__ ab) {
  const size_t idx = (size_t)blockIdx.x * 256 + threadIdx.x;
  const int z = (int)(idx & 1);
  out[idx] = ab[z] * out[idx] + ab[2 + z];
}

// ---------------------------------------------------------------------------
extern "C" void kernel_launch(void* const* d_in, const int* in_sizes, int n_in,
                              void* d_out, int out_size, void* d_ws, size_t ws_size,
                              hipStream_t stream) {
  const float* x     = (const float*)d_in[0];
  const float* w     = (const float*)d_in[1];
  const float* sw    = (const float*)d_in[2];
  const float* sb    = (const float*)d_in[3];
  const float* gamma = (const float*)d_in[4];
  const float* beta  = (const float*)d_in[5];
  const float* grid  = (const float*)d_in[6];
  const float* cay   = (const float*)d_in[7];
  float* out = (float*)d_out;

  char* ws = (char*)d_ws;
  float* bfrag   = (float*)(ws);                         // 8,650,752 B
  float* biassum = (float*)(ws + 8650752);               // 1,024 B
  float* part    = (float*)(ws + 8651776);               // 1,024 B
  float* ab      = (float*)(ws + 8652800);               // 16 B

  kan_pack <<<BF_FLOATS / 256, 256, 0, stream>>>(w, sw, cay, bfrag);
  kan_bias <<<1, 256, 0, stream>>>(sb, biassum);
  kan_gemm <<<32, 128, 0, stream>>>(x, grid, bfrag, biassum, out);
  kan_red  <<<64, 256, 0, stream>>>(out, part);
  kan_fin  <<<1, 64, 0, stream>>>(part, gamma, beta, ab);
  kan_apply<<<2048, 256, 0, stream>>>(out, ab);
}